// Model_2619930051268
// MI455X (gfx1250) — compile-verified
//
#include <hip/hip_runtime.h>
#include <math.h>

// Problem constants (match reference)
#define ALPHA_C 1.702f
#define LIMIT_C 7.0f
constexpr int B_TOK = 128;
constexpr int E_TOP = 4;
constexpr int NEXP  = 32;
constexpr int K_DIM = 512;   // d_model
constexpr int C_DIM = 1024;  // interleaved channels
constexpr int PAIRS = B_TOK * E_TOP;   // 512

// Tiling
constexpr int MB  = 32;          // token rows per block (2 stacked 16-row WMMA tiles)
constexpr int CB  = 128;         // channels per block (8 waves x 16)
constexpr int KC  = 32;          // K-chunk staged in LDS per iteration
constexpr int ST  = KC + 4;      // padded chunk row stride (dwords); 36*ln mod 64 distinct
constexpr int NCH = K_DIM / KC;  // 16 chunks

typedef float v2f __attribute__((ext_vector_type(2)));
typedef float v4f __attribute__((ext_vector_type(4)));
typedef float v8f __attribute__((ext_vector_type(8)));

// ---------------------------------------------------------------------------
// Kernel 1: bucket (token, slot) pairs by expert.
// ---------------------------------------------------------------------------
__global__ void moe_build_lists(const int* __restrict__ idx,
                                int* __restrict__ counts,
                                int* __restrict__ list) {
  __shared__ int scnt[NEXP];
  const int t = threadIdx.x;
  if (t < NEXP) scnt[t] = 0;
  __syncthreads();
  if (t < PAIRS) {
    const int e = idx[t];
    const int pos = atomicAdd(&scnt[e], 1);
    list[e * PAIRS + pos] = t;
  }
  __syncthreads();
  if (t < NEXP) counts[t] = scnt[t];
}

// ---------------------------------------------------------------------------
// Kernel 2: per-expert GEMM (32 tokens x 128 channels per block) with f32
// WMMA, double-buffered LDS chunks, fused bias/GLU epilogue.
// grid = (C/CB, PAIRS/MB, NEXP); block = 256 (8 waves)
// ---------------------------------------------------------------------------
__global__ __launch_bounds__(256)
void moe_mlp1_wmma(const float* __restrict__ x,      // [B,K]
                   const float* __restrict__ w,      // [NEXP,C,K]
                   const float* __restrict__ bias,   // [NEXP,C]
                   const int* __restrict__ counts,
                   const int* __restrict__ list,
                   float* __restrict__ out)          // [PAIRS, C/2]
{
  __shared__ float sX[2][MB * ST];   // gathered token rows, one K-chunk
  __shared__ float sW[2][CB * ST];   // weight K-chunk
  __shared__ int   sPid[MB];

  const int e     = blockIdx.z;
  const int cnt   = counts[e];
  const int tile0 = blockIdx.y * MB;
  if (tile0 >= cnt) return;
  const int cblk = blockIdx.x * CB;
  const int tid  = threadIdx.x;

  if (tid < MB) {
    const int r = tile0 + tid;
    sPid[tid] = (r < cnt) ? list[e * PAIRS + r] : -1;
  }
  __syncthreads();

  // Per-thread staging coordinates (KC/4 == 8 float4 per row)
  const int xrow = tid >> 3;            // 0..31
  const int xc4  = tid & 7;             // 0..7
  const int xpid = sPid[xrow];
  const int xtok = (xpid >= 0) ? (xpid >> 2) : 0;    // E_TOP == 4
  const float* xg = x + (size_t)xtok * K_DIM + xc4 * 4;

  const size_t wbase = (size_t)e * C_DIM * K_DIM + (size_t)cblk * K_DIM;
  // W: 4 float4 per thread; row = (tid + j*256)/8, c4 = tid&7
  const int wr0 = tid >> 3;             // rows wr0, wr0+32, wr0+64, wr0+96
  const float* wg = w + wbase + (size_t)wr0 * K_DIM + xc4 * 4;

  // ---- preload chunk 0 into buffer 0 ----
  {
    v4f rx = *(const v4f*)(xg);
    v4f rw0 = *(const v4f*)(wg);
    v4f rw1 = *(const v4f*)(wg + 32 * K_DIM);
    v4f rw2 = *(const v4f*)(wg + 64 * K_DIM);
    v4f rw3 = *(const v4f*)(wg + 96 * K_DIM);
    *(v4f*)(&sX[0][xrow * ST + xc4 * 4]) = rx;
    *(v4f*)(&sW[0][(wr0      ) * ST + xc4 * 4]) = rw0;
    *(v4f*)(&sW[0][(wr0 + 32 ) * ST + xc4 * 4]) = rw1;
    *(v4f*)(&sW[0][(wr0 + 64 ) * ST + xc4 * 4]) = rw2;
    *(v4f*)(&sW[0][(wr0 + 96 ) * ST + xc4 * 4]) = rw3;
  }

  const int wave = tid >> 5;
  const int lane = tid & 31;
  const int hi   = lane >> 4;      // selects K pair (0,1) vs (2,3)
  const int ln   = lane & 15;

  v8f acc0 = {0.f, 0.f, 0.f, 0.f, 0.f, 0.f, 0.f, 0.f};  // tokens 0..15
  v8f acc1 = {0.f, 0.f, 0.f, 0.f, 0.f, 0.f, 0.f, 0.f};  // tokens 16..31

  int buf = 0;
#pragma unroll 2
  for (int ic = 0; ic < NCH; ++ic) {
    __syncthreads();   // buf's stores (from previous iter) visible

    // Issue next chunk's global loads early (latency hidden behind WMMAs)
    const bool hasNext = (ic + 1 < NCH);
    const int  kn = (ic + 1) * KC;
    v4f rx = {}, rw0 = {}, rw1 = {}, rw2 = {}, rw3 = {};
    if (hasNext) {
      rx  = *(const v4f*)(xg + kn);
      rw0 = *(const v4f*)(wg + kn);
      rw1 = *(const v4f*)(wg + 32 * K_DIM + kn);
      rw2 = *(const v4f*)(wg + 64 * K_DIM + kn);
      rw3 = *(const v4f*)(wg + 96 * K_DIM + kn);
    }

    // Compute on current buffer: 8 k-steps x 2 stacked tiles = 16 WMMAs
    const float* a0p = &sX[buf][ln * ST + 2 * hi];
    const float* a1p = &sX[buf][(16 + ln) * ST + 2 * hi];
    const float* bp  = &sW[buf][(wave * 16 + ln) * ST + 2 * hi];
#pragma unroll
    for (int k = 0; k < KC; k += 4) {
      v2f b  = *(const v2f*)(bp + k);
      v2f a0 = *(const v2f*)(a0p + k);
      v2f a1 = *(const v2f*)(a1p + k);
      acc0 = __builtin_amdgcn_wmma_f32_16x16x4_f32(false, a0, false, b,
                                                   (short)0, acc0, false, false);
      acc1 = __builtin_amdgcn_wmma_f32_16x16x4_f32(false, a1, false, b,
                                                   (short)0, acc1, false, false);
    }

    // Store next chunk into the other buffer
    if (hasNext) {
      const int nb = buf ^ 1;
      *(v4f*)(&sX[nb][xrow * ST + xc4 * 4]) = rx;
      *(v4f*)(&sW[nb][(wr0      ) * ST + xc4 * 4]) = rw0;
      *(v4f*)(&sW[nb][(wr0 + 32 ) * ST + xc4 * 4]) = rw1;
      *(v4f*)(&sW[nb][(wr0 + 64 ) * ST + xc4 * 4]) = rw2;
      *(v4f*)(&sW[nb][(wr0 + 96 ) * ST + xc4 * 4]) = rw3;
    }
    buf ^= 1;
  }

  // Epilogue: D layout -> VGPR r holds (M = r + 8*hi, N = ln).
  // Channel parity == lane parity; even lane = glu, odd lane = linear.
  const int   c  = cblk + wave * 16 + ln;
  const float bv = bias[e * C_DIM + c];

#pragma unroll
  for (int half = 0; half < 2; ++half) {
    const v8f& acc = half ? acc1 : acc0;
#pragma unroll
    for (int r = 0; r < 8; ++r) {
      const int m = half * 16 + r + 8 * hi;
      float t = acc[r] + bv;
      float other = __shfl_xor(t, 1, 32);   // partner channel (wave32)
      if ((ln & 1) == 0) {
        const float xgv = fminf(t, LIMIT_C);
        const float xlv = fminf(fmaxf(other, -LIMIT_C), LIMIT_C);
        const float g   = xgv * (1.0f / (1.0f + __expf(-ALPHA_C * xgv)));
        const float o   = g * (xlv + 1.0f);
        const int row = tile0 + m;
        if (row < cnt) {
          const int pid = sPid[m];
          out[(size_t)pid * (C_DIM / 2) + (c >> 1)] = o;
        }
      }
    }
  }
}

// ---------------------------------------------------------------------------
extern "C" void kernel_launch(void* const* d_in, const int* in_sizes, int n_in,
                              void* d_out, int out_size, void* d_ws, size_t ws_size,
                              hipStream_t stream) {
  (void)in_sizes; (void)n_in; (void)out_size; (void)ws_size;
  const float* normed_x    = (const float*)d_in[0];
  const int*   expert_idx  = (const int*)d_in[1];
  const float* mlp1_weight = (const float*)d_in[2];
  const float* mlp1_bias   = (const float*)d_in[3];
  float*       out         = (float*)d_out;

  int* counts = (int*)d_ws;            // 32 ints
  int* list   = counts + NEXP;         // 32 * 512 ints

  moe_build_lists<<<1, PAIRS, 0, stream>>>(expert_idx, counts, list);

  dim3 grid(C_DIM / CB, PAIRS / MB, NEXP);   // (8, 16, 32)
  moe_mlp1_wmma<<<grid, 256, 0, stream>>>(normed_x, mlp1_weight, mlp1_bias,
                                          counts, list, out);
}